// SparseAttentionDemo_14396730376894
// MI455X (gfx1250) — compile-verified
//
#include <hip/hip_runtime.h>

typedef __attribute__((ext_vector_type(2))) float v2f;
typedef __attribute__((ext_vector_type(4))) float v4f;
typedef __attribute__((ext_vector_type(8))) float v8f;

#define BQ   4
#define SEQ  4096
#define DM   1024
#define HID  64
#define KTOP 204   // int(4096 * 0.05)

// ---------------------------------------------------------------------------
// Stage 1: scores[b,s] = relu(emb[b,s,:] @ W1 + b1) @ W2 + b2
// One wave -> 16 rows. 4 accumulators (N=0..63), K loop of 256 x (16x16x4).
// A layout (16x4 f32): lanes 0-15 hold M=lane, VGPR0/1 = K0/K1; lanes 16-31 = K2/K3.
// B layout (4x16 f32): lanes 0-15 hold N=lane, VGPR0/1 = K0/K1; lanes 16-31 = K2/K3.
// C/D layout: VGPR v, lane l -> M = v + 8*(l>>4), N = l&15.
// ---------------------------------------------------------------------------
__global__ __launch_bounds__(128) void mlp_scores_kernel(
    const float* __restrict__ emb, const float* __restrict__ W1,
    const float* __restrict__ b1,  const float* __restrict__ W2,
    const float* __restrict__ b2,  float* __restrict__ scores)
{
    const int wave = threadIdx.x >> 5;
    const int lane = threadIdx.x & 31;
    const int half = lane >> 4;     // 0 or 1
    const int li   = lane & 15;
    const int m0   = (blockIdx.x * 4 + wave) * 16;

    const float* arow = emb + (size_t)(m0 + li) * DM + half * 2;
    const float* w1p  = W1 + (half * 2) * HID + li;

    v8f acc0 = {}, acc1 = {}, acc2 = {}, acc3 = {};

    for (int k = 0; k < DM; k += 4) {
        v2f a = *(const v2f*)(arow + k);          // E[m][k+2h], E[m][k+2h+1]
        const float* wk = w1p + k * HID;
        v2f bv0, bv1, bv2, bv3;
        bv0.x = wk[0];   bv0.y = wk[HID + 0];
        bv1.x = wk[16];  bv1.y = wk[HID + 16];
        bv2.x = wk[32];  bv2.y = wk[HID + 32];
        bv3.x = wk[48];  bv3.y = wk[HID + 48];
        acc0 = __builtin_amdgcn_wmma_f32_16x16x4_f32(false, a, false, bv0, (short)0, acc0, false, false);
        acc1 = __builtin_amdgcn_wmma_f32_16x16x4_f32(false, a, false, bv1, (short)0, acc1, false, false);
        acc2 = __builtin_amdgcn_wmma_f32_16x16x4_f32(false, a, false, bv2, (short)0, acc2, false, false);
        acc3 = __builtin_amdgcn_wmma_f32_16x16x4_f32(false, a, false, bv3, (short)0, acc3, false, false);
    }

    // Epilogue: relu(h + b1) dot W2, reduce over the 16 lanes of each half.
    const float b1a = b1[li],      b1b = b1[16 + li];
    const float b1c = b1[32 + li], b1d = b1[48 + li];
    const float w2a = W2[li],      w2b = W2[16 + li];
    const float w2c = W2[32 + li], w2d = W2[48 + li];
    const float b2s = b2[0];

    #pragma unroll
    for (int v = 0; v < 8; ++v) {
        float p = 0.0f;
        p += fmaxf(acc0[v] + b1a, 0.0f) * w2a;
        p += fmaxf(acc1[v] + b1b, 0.0f) * w2b;
        p += fmaxf(acc2[v] + b1c, 0.0f) * w2c;
        p += fmaxf(acc3[v] + b1d, 0.0f) * w2d;
        // reduce within each 16-lane half (masks 1,2,4,8 stay inside the half)
        p += __shfl_xor(p, 1, 32);
        p += __shfl_xor(p, 2, 32);
        p += __shfl_xor(p, 4, 32);
        p += __shfl_xor(p, 8, 32);
        if (li == 0)
            scores[m0 + v + 8 * half] = p + b2s;
    }
}

// ---------------------------------------------------------------------------
// Stage 2: exact top-k by rank counting (matches lax.top_k: descending,
// ties broken by lower index). One block per batch; scores in LDS.
// mask[b][i] = (rank<k) ? 1/k : 0 ; idx_out[b*k + rank] = float(i).
// ---------------------------------------------------------------------------
__global__ __launch_bounds__(256) void topk_mask_kernel(
    const float* __restrict__ scores, float* __restrict__ mask,
    float* __restrict__ idx_out)
{
    __shared__ float s[SEQ];
    const int b   = blockIdx.x;
    const int tid = threadIdx.x;

    for (int i = tid; i < SEQ; i += 256) s[i] = scores[b * SEQ + i];
    __syncthreads();

    float my[16];
    int   rank[16];
    #pragma unroll
    for (int u = 0; u < 16; ++u) { my[u] = s[tid + u * 256]; rank[u] = 0; }

    for (int j = 0; j < SEQ; ++j) {
        const float sj = s[j];   // broadcast read, conflict-free
        #pragma unroll
        for (int u = 0; u < 16; ++u) {
            const int i = tid + u * 256;
            rank[u] += (sj > my[u]) || (sj == my[u] && j < i);
        }
    }

    const float invk = 1.0f / (float)KTOP;
    #pragma unroll
    for (int u = 0; u < 16; ++u) {
        const int  i   = tid + u * 256;
        const bool sel = rank[u] < KTOP;
        mask[b * SEQ + i] = sel ? invk : 0.0f;
        if (sel) idx_out[b * KTOP + rank[u]] = (float)i;
    }
}

// ---------------------------------------------------------------------------
// Stage 3: out[b][i][j] = mask[b][j]  (268 MB of streaming stores -> NT hint).
// Each block caches one mask row in registers and writes 8 output rows.
// ---------------------------------------------------------------------------
__global__ __launch_bounds__(256) void broadcast_kernel(
    const float* __restrict__ mask, float* __restrict__ out)
{
    const int ROWS = 8;
    const int b    = blockIdx.x / (SEQ / ROWS);
    const int r0   = (blockIdx.x % (SEQ / ROWS)) * ROWS;
    const int tid  = threadIdx.x;

    const v4f* m4 = (const v4f*)(mask + (size_t)b * SEQ);
    v4f vals[4];
    #pragma unroll
    for (int c = 0; c < 4; ++c) vals[c] = m4[tid + c * 256];

    v4f* o4 = (v4f*)(out + ((size_t)(b * SEQ + r0)) * SEQ);
    #pragma unroll
    for (int r = 0; r < ROWS; ++r) {
        #pragma unroll
        for (int c = 0; c < 4; ++c)
            __builtin_nontemporal_store(vals[c],
                o4 + (size_t)r * (SEQ / 4) + tid + c * 256);
    }
}

extern "C" void kernel_launch(void* const* d_in, const int* in_sizes, int n_in,
                              void* d_out, int out_size, void* d_ws, size_t ws_size,
                              hipStream_t stream)
{
    const float* emb = (const float*)d_in[0];
    const float* W1  = (const float*)d_in[1];
    const float* b1  = (const float*)d_in[2];
    const float* W2  = (const float*)d_in[3];
    const float* b2  = (const float*)d_in[4];

    float* out     = (float*)d_out;
    float* idx_out = out + (size_t)BQ * SEQ * SEQ;   // indices tail (as float)

    float* scores = (float*)d_ws;                    // BQ*SEQ floats
    float* mask   = scores + BQ * SEQ;               // BQ*SEQ floats

    // 16384 rows / (4 waves * 16 rows) = 256 blocks
    mlp_scores_kernel<<<(BQ * SEQ) / 64, 128, 0, stream>>>(emb, W1, b1, W2, b2, scores);
    topk_mask_kernel<<<BQ, 256, 0, stream>>>(scores, mask, idx_out);
    broadcast_kernel<<<BQ * (SEQ / 8), 256, 0, stream>>>(mask, out);
}